// MotifClusteringPool_26388279066709
// MI455X (gfx1250) — compile-verified
//
#include <hip/hip_runtime.h>
#include <hip/hip_bf16.h>

// MotifClusteringPool for MI455X (gfx1250), fp32, wave32.
// Memory-bound: ~600MB of N x N traffic dominates (23.3 TB/s => ~26us floor).
// WMMA F32 16x16x4 everywhere the tile loads are contiguous:
//  - transposed aggregation M^T Y (A-tiles row-contiguous in motif)
//  - dense 128-wide GEMMs (x@Wg, relu(h@W1+b1), H1@W2 + fused softmax)
//  - motif@S / adjacency@S via LDS-staged tiles (coalesced b128 stream,
//    conflict-free ds_load_b64 A-fragments, 2 WMMAs per k-step)
//  - S^T pooling with K split across blocks + global_atomic_add_f32

#define HID 128
#define CK  16

typedef __attribute__((ext_vector_type(2))) float v2f;
typedef __attribute__((ext_vector_type(8))) float v8f;

__device__ __forceinline__ v8f wmma4(v2f a, v2f b, v8f c) {
  // D = A(16x4) * B(4x16) + C(16x16), fp32
  return __builtin_amdgcn_wmma_f32_16x16x4_f32(false, a, false, b, (short)0, c,
                                               false, false);
}

// ---------------------------------------------------------------- zero init
__global__ void zero_kernel(float* __restrict__ dout, float* __restrict__ deg,
                            float* __restrict__ StS, float* __restrict__ scal,
                            int n) {
  int t = blockIdx.x * blockDim.x + threadIdx.x;
  if (t < 2560) dout[t] = 0.0f;      // X_pool + A_pool + motif_pool accumulators
  if (t < n)    deg[t]  = 0.0f;
  if (t < 256)  StS[t]  = 0.0f;
  if (t < 2)    scal[t] = 0.0f;
}

// ------------------------------------------------- column sums of motif (deg)
__global__ void colsum_kernel(const float* __restrict__ M,
                              float* __restrict__ deg, int n, int rows_per_blk) {
  int col = blockIdx.x * blockDim.x + threadIdx.x;  // coalesced across row
  int r0  = blockIdx.y * rows_per_blk;
  float s = 0.0f;
  for (int r = r0; r < r0 + rows_per_blk; ++r)
    s += M[(long)r * n + col];
  atomicAdd(&deg[col], s);
}

__global__ void dinv_kernel(const float* __restrict__ deg,
                            float* __restrict__ dinv, int n) {
  int t = blockIdx.x * blockDim.x + threadIdx.x;
  if (t < n) dinv[t] = rsqrtf(deg[t] + 1.0f);  // +1: self loop; always > 0
}

// ------------------------------------- generic [*,128]@[128,128] WMMA GEMM
// out = act(A @ B + bias); one wave per 16x16 tile; 8 waves cover HID=128.
// Compile-time specialization keeps the epilogue branch-free.
template <bool HAS_BIAS, bool RELU>
__global__ void gemm_nn128(const float* __restrict__ A, const float* __restrict__ B,
                           const float* __restrict__ bias, float* __restrict__ out) {
  int wave = threadIdx.x >> 5;
  int lane = threadIdx.x & 31;
  int half = lane >> 4, l16 = lane & 15;
  int r0 = blockIdx.x * 16;
  int n0 = wave * 16;
  v8f acc = {0.f, 0.f, 0.f, 0.f, 0.f, 0.f, 0.f, 0.f};
#pragma unroll 4
  for (int k0 = 0; k0 < HID; k0 += 4) {
    int ka = k0 + 2 * half;
    v2f a, b;
    a[0] = A[(r0 + l16) * HID + ka];
    a[1] = A[(r0 + l16) * HID + ka + 1];
    b[0] = B[ka * HID + n0 + l16];
    b[1] = B[(ka + 1) * HID + n0 + l16];
    acc = wmma4(a, b, acc);
  }
  float bv = HAS_BIAS ? bias[n0 + l16] : 0.0f;
#pragma unroll
  for (int r = 0; r < 8; ++r) {
    int row = r0 + r + 8 * half;
    float v = acc[r] + bv;
    if (RELU) v = fmaxf(v, 0.0f);
    out[row * HID + n0 + l16] = v;
  }
}

// ------------------------- h = tanh(dinv_i * (M^T (dinv (.) XW))_i + b_gcn)
// M = motif + I. Transposed access -> A-tiles are row-contiguous in motif.
// Wave owns a 16-col stripe of M; 8 acc tiles reuse each A fragment 8x.
__global__ void agg_kernel(const float* __restrict__ motif,
                           const float* __restrict__ XW,
                           const float* __restrict__ dinv,
                           const float* __restrict__ bgcn,
                           float* __restrict__ h, int n) {
  int wave = threadIdx.x >> 5;
  int lane = threadIdx.x & 31;
  int half = lane >> 4, l16 = lane & 15;
  int i0 = blockIdx.x * 128 + wave * 16;

  v8f acc[8];
#pragma unroll
  for (int t = 0; t < 8; ++t) acc[t] = (v8f){0.f,0.f,0.f,0.f,0.f,0.f,0.f,0.f};

  for (int k0 = 0; k0 < n; k0 += 4) {
    int ka = k0 + 2 * half;
    v2f a;
    a[0] = motif[(long)ka * n + i0 + l16];        // 64B contiguous per half-wave
    a[1] = motif[(long)(ka + 1) * n + i0 + l16];
    float d0 = dinv[ka], d1 = dinv[ka + 1];
#pragma unroll
    for (int t = 0; t < 8; ++t) {
      v2f b;
      b[0] = d0 * XW[ka * HID + t * 16 + l16];    // Y = dinv (.) XW on the fly
      b[1] = d1 * XW[(ka + 1) * HID + t * 16 + l16];
      acc[t] = wmma4(a, b, acc[t]);
    }
  }
#pragma unroll
  for (int r = 0; r < 8; ++r) {
    int row = i0 + r + 8 * half;
    float di = dinv[row];
#pragma unroll
    for (int t = 0; t < 8; ++t) {
      int col = t * 16 + l16;
      float T = acc[t][r] + di * XW[row * HID + col];  // + I self-loop term
      h[row * HID + col] = tanhf(di * T + bgcn[col]);
    }
  }
}

// -------------------- logits = H1 @ W2 + b2, fused 16-lane-group softmax -> S
__global__ void logits_softmax(const float* __restrict__ H1,
                               const float* __restrict__ W2,
                               const float* __restrict__ b2,
                               float* __restrict__ S) {
  int wave = threadIdx.x >> 5;
  int lane = threadIdx.x & 31;
  int half = lane >> 4, l16 = lane & 15;
  int i0 = blockIdx.x * 128 + wave * 16;
  v8f acc = {0.f, 0.f, 0.f, 0.f, 0.f, 0.f, 0.f, 0.f};
#pragma unroll 4
  for (int k0 = 0; k0 < HID; k0 += 4) {
    int ka = k0 + 2 * half;
    v2f a, b;
    a[0] = H1[(i0 + l16) * HID + ka];
    a[1] = H1[(i0 + l16) * HID + ka + 1];
    b[0] = W2[ka * CK + l16];
    b[1] = W2[(ka + 1) * CK + l16];
    acc = wmma4(a, b, acc);
  }
#pragma unroll
  for (int r = 0; r < 8; ++r) {
    float v = acc[r] + b2[l16];
    float m = v;  // row's 16 logits live in one VGPR across a 16-lane group
    for (int mask = 1; mask < 16; mask <<= 1)
      m = fmaxf(m, __shfl_xor(m, mask, 32));
    float e = expf(v - m);
    float s = e;
    for (int mask = 1; mask < 16; mask <<= 1)
      s += __shfl_xor(s, mask, 32);
    S[(i0 + r + 8 * half) * CK + l16] = e / s;
  }
}

// ---------------- MS = motif@S, AS = adjacency@S, deg_m row sums (one pass)
// LDS-staged WMMA: block stages 128x32 tiles of both matrices with coalesced
// b128 loads (64B/thread), waves read A-fragments as conflict-free ds_load_b64
// (row stride 36 floats => 16 distinct banks), 2 WMMAs per k-step.
__global__ void bigAS_kernel(const float* __restrict__ motif,
                             const float* __restrict__ adj,
                             const float* __restrict__ S,
                             float* __restrict__ MS, float* __restrict__ AS,
                             float* __restrict__ degm, int n) {
  __shared__ float smot[128 * 36];
  __shared__ float sadj[128 * 36];
  const int tid  = threadIdx.x;
  const int wave = tid >> 5, lane = tid & 31;
  const int half = lane >> 4, l16 = lane & 15;
  const int i0blk = blockIdx.x * 128;
  const int i0w   = i0blk + wave * 16;

  // staging coords: 2 threads per row, 16 contiguous floats (64B) each
  const int srow = tid >> 1;
  const int scol = (tid & 1) * 16;
  const float* mrow = motif + (long)(i0blk + srow) * n + scol;
  const float* arow = adj   + (long)(i0blk + srow) * n + scol;
  float* smp = &smot[srow * 36 + scol];
  float* sap = &sadj[srow * 36 + scol];

  v8f accm = {0.f, 0.f, 0.f, 0.f, 0.f, 0.f, 0.f, 0.f};
  v8f acca = {0.f, 0.f, 0.f, 0.f, 0.f, 0.f, 0.f, 0.f};
  float dm = 0.f;

  for (int kc = 0; kc < n; kc += 32) {
#pragma unroll
    for (int q = 0; q < 4; ++q) {
      *(float4*)(smp + q * 4) = *(const float4*)(mrow + kc + q * 4);
      *(float4*)(sap + q * 4) = *(const float4*)(arow + kc + q * 4);
    }
    if (kc + 32 < n) {  // hint the next chunk of both streams
      __builtin_prefetch(mrow + kc + 32, 0, 0);
      __builtin_prefetch(arow + kc + 32, 0, 0);
    }
    __syncthreads();
#pragma unroll
    for (int kk = 0; kk < 32; kk += 4) {
      int ka = kk + 2 * half;
      v2f am = *(const v2f*)&smot[(wave * 16 + l16) * 36 + ka];
      v2f aa = *(const v2f*)&sadj[(wave * 16 + l16) * 36 + ka];
      int kg = kc + ka;
      v2f b;
      b[0] = S[kg * CK + l16];        // L2-resident, 64B contiguous
      b[1] = S[(kg + 1) * CK + l16];
      accm = wmma4(am, b, accm);
      acca = wmma4(aa, b, acca);
      dm += am[0] + am[1];            // row-sum of motif falls out for free
    }
    __syncthreads();
  }
  dm += __shfl_xor(dm, 16, 32);       // merge the two k-half subsets
  if (lane < 16) degm[i0w + l16] = dm;
#pragma unroll
  for (int r = 0; r < 8; ++r) {
    int row = i0w + r + 8 * half;
    MS[row * CK + l16] = accm[r];
    AS[row * CK + l16] = acca[r];
  }
}

// ------------- pooling: X_pool=S^T h, A_pool=S^T AS, motif_pool=S^T MS,
// StS=S^T S, cut_num/cut_den partials. K split across blocks, atomic reduce.
__global__ void pool_kernel(const float* __restrict__ S, const float* __restrict__ h,
                            const float* __restrict__ MS, const float* __restrict__ AS,
                            const float* __restrict__ degm,
                            float* __restrict__ Xpool, float* __restrict__ Apool,
                            float* __restrict__ Mpool, float* __restrict__ StS,
                            float* __restrict__ scal) {
  int wave = threadIdx.x >> 5;
  int lane = threadIdx.x & 31;
  int half = lane >> 4, l16 = lane & 15;
  int k0blk = blockIdx.x * 128;

  if (wave < 11) {
    const float* Bsrc;
    float* dst;
    int ldb, ldd, n0;
    if (wave < 8)       { Bsrc = h;  ldb = HID; n0 = wave * 16; dst = Xpool; ldd = HID; }
    else if (wave == 8) { Bsrc = AS; ldb = CK;  n0 = 0;         dst = Apool; ldd = CK;  }
    else if (wave == 9) { Bsrc = MS; ldb = CK;  n0 = 0;         dst = Mpool; ldd = CK;  }
    else                { Bsrc = S;  ldb = CK;  n0 = 0;         dst = StS;   ldd = CK;  }
    v8f acc = {0.f, 0.f, 0.f, 0.f, 0.f, 0.f, 0.f, 0.f};
    for (int k0 = k0blk; k0 < k0blk + 128; k0 += 4) {
      int ka = k0 + 2 * half;
      v2f a, b;
      a[0] = S[ka * CK + l16];            // S^T A-tile: contiguous rows of S
      a[1] = S[(ka + 1) * CK + l16];
      b[0] = Bsrc[ka * ldb + n0 + l16];
      b[1] = Bsrc[(ka + 1) * ldb + n0 + l16];
      acc = wmma4(a, b, acc);
    }
#pragma unroll
    for (int r = 0; r < 8; ++r)
      atomicAdd(&dst[(r + 8 * half) * ldd + n0 + l16], acc[r]);
  } else {
    // scalar partials: cut_num = sum(MS*S), cut_den = sum(deg_m * ||S_i||^2)
    float pn = 0.f, pd = 0.f;
    for (int k = k0blk + lane; k < k0blk + 128; k += 32) {
      float sq = 0.f, cn = 0.f;
#pragma unroll
      for (int c = 0; c < CK; ++c) {
        float s = S[k * CK + c];
        sq += s * s;
        cn += MS[k * CK + c] * s;
      }
      pn += cn;
      pd += degm[k] * sq;
    }
    for (int mask = 16; mask >= 1; mask >>= 1) {
      pn += __shfl_xor(pn, mask, 32);
      pd += __shfl_xor(pd, mask, 32);
    }
    if (lane == 0) {
      atomicAdd(&scal[0], pn);
      atomicAdd(&scal[1], pd);
    }
  }
}

// ----------------------------- finalize: losses + zero pooled-matrix diagonals
__global__ void finalize_kernel(float* __restrict__ dout,
                                const float* __restrict__ StS,
                                const float* __restrict__ scal, int n) {
  __shared__ float red[256];
  int t = threadIdx.x;
  float s = StS[t];
  float d = s - (((t % 17) == 0) ? 1.0f : 0.0f);  // diag of 16x16: idx 0,17,...
  red[t] = d * d;
  __syncthreads();
  for (int w = 128; w > 0; w >>= 1) {
    if (t < w) red[t] += red[t + w];
    __syncthreads();
  }
  if (t == 0) {
    long base = 2560 + (long)n * CK;
    dout[base]     = -scal[0] / (scal[1] + 1e-8f);  // L_cut
    dout[base + 1] = sqrtf(red[0]) / 4.0f;          // L_ortho (sqrt(16)=4)
  }
  if (t < CK) {
    dout[2048 + t * 17] = 0.0f;  // A_pool diag
    dout[2304 + t * 17] = 0.0f;  // motif_pool diag
  }
}

extern "C" void kernel_launch(void* const* d_in, const int* in_sizes, int n_in,
                              void* d_out, int out_size, void* d_ws, size_t ws_size,
                              hipStream_t stream) {
  (void)n_in; (void)out_size; (void)ws_size;
  const float* x     = (const float*)d_in[0];
  const float* adj   = (const float*)d_in[1];
  const float* motif = (const float*)d_in[2];
  const float* Wg    = (const float*)d_in[3];
  const float* bg    = (const float*)d_in[4];
  const float* W1    = (const float*)d_in[5];
  const float* b1    = (const float*)d_in[6];
  const float* W2    = (const float*)d_in[7];
  const float* b2    = (const float*)d_in[8];
  float* out = (float*)d_out;
  const int n = in_sizes[0] / HID;  // 6144

  // workspace carve-up (~10.3 MB of fp32)
  float* ws = (float*)d_ws;
  size_t o = 0;
  float* XW   = ws + o; o += (size_t)n * HID;
  float* deg  = ws + o; o += n;
  float* dinv = ws + o; o += n;
  float* degm = ws + o; o += n;
  float* hbuf = ws + o; o += (size_t)n * HID;
  float* H1   = ws + o; o += (size_t)n * HID;
  float* MS   = ws + o; o += (size_t)n * CK;
  float* AS   = ws + o; o += (size_t)n * CK;
  float* StS  = ws + o; o += 256;
  float* scal = ws + o; o += 2;

  float* Xpool = out;                 // [16,128]
  float* Apool = out + 2048;          // [16,16]
  float* Mpool = out + 2304;          // [16,16]
  float* Sm    = out + 2560;          // [n,16]

  zero_kernel<<<(n + 255) / 256, 256, 0, stream>>>(out, deg, StS, scal, n);
  colsum_kernel<<<dim3(n / 256, 48), 256, 0, stream>>>(motif, deg, n, n / 48);
  dinv_kernel<<<(n + 255) / 256, 256, 0, stream>>>(deg, dinv, n);
  gemm_nn128<false, false><<<n / 16, 256, 0, stream>>>(x, Wg, nullptr, XW);
  agg_kernel<<<n / 128, 256, 0, stream>>>(motif, XW, dinv, bg, hbuf, n);
  gemm_nn128<true, true><<<n / 16, 256, 0, stream>>>(hbuf, W1, b1, H1);
  logits_softmax<<<n / 128, 256, 0, stream>>>(H1, W2, b2, Sm);
  bigAS_kernel<<<n / 128, 256, 0, stream>>>(motif, adj, Sm, MS, AS, degm, n);
  pool_kernel<<<n / 128, 384, 0, stream>>>(Sm, hbuf, MS, AS, degm,
                                           Xpool, Apool, Mpool, StS, scal);
  finalize_kernel<<<1, 256, 0, stream>>>(out, StS, scal, n);
}